// PyramidROIAlign_1202590842975
// MI455X (gfx1250) — compile-verified
//
#include <hip/hip_runtime.h>
#include <math.h>

#define POOL 7
#define CH 256
#define WAVES 8   // 256 threads / wave32

typedef float __attribute__((ext_vector_type(4))) f4;
typedef int   __attribute__((ext_vector_type(4))) v4i;

// ---- CDNA5 async global->LDS path (guarded; fallback = direct loads) ------
#if defined(__has_builtin)
#if __has_builtin(__builtin_amdgcn_global_load_async_to_lds_b128)
#define HAVE_ASYNC_LDS 1
#endif
#endif

#if defined(HAVE_ASYNC_LDS)
typedef __attribute__((address_space(1))) v4i* gv4i_p;   // global int4*
typedef __attribute__((address_space(3))) v4i* lv4i_p;   // lds    int4*

__device__ __forceinline__ void async_ld16(const float* g, float* l) {
  // (global src, lds dst, imm offset, imm cpol)
  __builtin_amdgcn_global_load_async_to_lds_b128(
      (gv4i_p)(void*)g, (lv4i_p)(void*)l, 0, 0);
}
#if __has_builtin(__builtin_amdgcn_s_wait_asynccnt)
#define WAIT_ASYNC() do { __builtin_amdgcn_s_wait_asynccnt(0); \
                          asm volatile("" ::: "memory"); } while (0)
#else
#define WAIT_ASYNC() asm volatile("s_wait_asynccnt 0" ::: "memory")
#endif
#endif

__device__ __forceinline__ f4 bilerp(f4 tl, f4 tr, f4 bl, f4 br, float wx, float wy) {
  f4 top = tl + (tr - tl) * wx;
  f4 bot = bl + (br - bl) * wx;
  return top + (bot - top) * wy;
}

__global__ __launch_bounds__(256) void roialign_kernel(
    const float* __restrict__ rois,   // [B,N,4]  (y1,x1,y2,x2) normalized
    const float* __restrict__ metas,  // [B,11]
    const float* __restrict__ fm2,    // [B,256,256,256]
    const float* __restrict__ fm3,    // [B,128,128,256]
    const float* __restrict__ fm4,    // [B, 64, 64,256]
    const float* __restrict__ fm5,    // [B, 32, 32,256]
    float* __restrict__ out,          // [B,N,7,7,256]
    int N)
{
#if defined(HAVE_ASYNC_LDS)
  __shared__ float smem[WAVES * 4 * CH];   // 32 KB: 4 corner vectors per wave
#endif
  const int roi  = blockIdx.x;             // b*N + n
  const int b    = roi / N;
  const int tid  = threadIdx.x;
  const int wave = tid >> 5;
  const int lane = tid & 31;

  // ---- box + level selection (uniform per block) --------------------------
  const float by1 = rois[roi * 4 + 0];
  const float bx1 = rois[roi * 4 + 1];
  const float by2 = rois[roi * 4 + 2];
  const float bx2 = rois[roi * 4 + 3];
  const float padh = metas[b * 11 + 7];
  const float padw = metas[b * 11 + 8];

  const float h = by2 - by1;
  const float w = bx2 - bx1;
  // roi_level = log2( sqrt(h*w) * sqrt(area) / 224 );  lvl = clip(4+round, 2, 5)
  const float rl = log2f(sqrtf(h * w) * sqrtf(padh * padw) / 224.0f);
  int lvl = 4 + (int)rintf(rl);            // rintf = round-half-even, matches jnp.round
  lvl = lvl < 2 ? 2 : (lvl > 5 ? 5 : lvl);

  const float* fm;
  int H;
  if      (lvl == 2) { fm = fm2; H = 256; }
  else if (lvl == 3) { fm = fm3; H = 128; }
  else if (lvl == 4) { fm = fm4; H = 64;  }
  else               { fm = fm5; H = 32;  }
  const int W = H;
  fm += (size_t)b * H * W * CH;

  const int c0 = lane * 4;                 // channels [c0, c0+4)
  const int c1 = 128 + lane * 4;           // channels [c1, c1+4)

#if defined(HAVE_ASYNC_LDS)
  float* buf = smem + wave * 4 * CH;       // [4][256] corner staging
#endif

  // ---- 49 sample points, 8 waves round-robin ------------------------------
  for (int p = wave; p < POOL * POOL; p += WAVES) {
    const int gi = p / POOL;
    const int gj = p % POOL;
    const float ys = (by1 + (float)gi * (1.0f / 6.0f) * h) * (float)(H - 1);
    const float xs = (bx1 + (float)gj * (1.0f / 6.0f) * w) * (float)(W - 1);
    const float y0f = floorf(ys), x0f = floorf(xs);
    const float wy = ys - y0f;
    const float wx = xs - x0f;
    int y0 = (int)y0f; y0 = y0 < 0 ? 0 : (y0 > H - 1 ? H - 1 : y0);
    int x0 = (int)x0f; x0 = x0 < 0 ? 0 : (x0 > W - 1 ? W - 1 : x0);
    const int y1i = y0 + 1 > H - 1 ? H - 1 : y0 + 1;
    const int x1i = x0 + 1 > W - 1 ? W - 1 : x0 + 1;

    const float* ptl  = fm + (size_t)(y0  * W + x0 ) * CH;
    const float* ptr_ = fm + (size_t)(y0  * W + x1i) * CH;
    const float* pbl  = fm + (size_t)(y1i * W + x0 ) * CH;
    const float* pbr  = fm + (size_t)(y1i * W + x1i) * CH;

    f4 tl0, tl1, tr0, tr1, bl0, bl1, br0, br1;
#if defined(HAVE_ASYNC_LDS)
    // Stage 4x1KB corner vectors into LDS: 8 async b128 ops (512B/wave each),
    // every lane copies contiguous 16B -> fully coalesced.
    async_ld16(ptl  + c0, buf + 0 * CH + c0);
    async_ld16(ptl  + c1, buf + 0 * CH + c1);
    async_ld16(ptr_ + c0, buf + 1 * CH + c0);
    async_ld16(ptr_ + c1, buf + 1 * CH + c1);
    async_ld16(pbl  + c0, buf + 2 * CH + c0);
    async_ld16(pbl  + c1, buf + 2 * CH + c1);
    async_ld16(pbr  + c0, buf + 3 * CH + c0);
    async_ld16(pbr  + c1, buf + 3 * CH + c1);
    WAIT_ASYNC();
    tl0 = *(const f4*)(buf + 0 * CH + c0);  tl1 = *(const f4*)(buf + 0 * CH + c1);
    tr0 = *(const f4*)(buf + 1 * CH + c0);  tr1 = *(const f4*)(buf + 1 * CH + c1);
    bl0 = *(const f4*)(buf + 2 * CH + c0);  bl1 = *(const f4*)(buf + 2 * CH + c1);
    br0 = *(const f4*)(buf + 3 * CH + c0);  br1 = *(const f4*)(buf + 3 * CH + c1);
#else
    tl0 = *(const f4*)(ptl  + c0);  tl1 = *(const f4*)(ptl  + c1);
    tr0 = *(const f4*)(ptr_ + c0);  tr1 = *(const f4*)(ptr_ + c1);
    bl0 = *(const f4*)(pbl  + c0);  bl1 = *(const f4*)(pbl  + c1);
    br0 = *(const f4*)(pbr  + c0);  br1 = *(const f4*)(pbr  + c1);
#endif

    const f4 r0 = bilerp(tl0, tr0, bl0, br0, wx, wy);
    const f4 r1 = bilerp(tl1, tr1, bl1, br1, wx, wy);

    float* o = out + ((size_t)roi * (POOL * POOL) + p) * CH;
    // Output is write-once streamed (100 MB): non-temporal b128 stores.
    __builtin_nontemporal_store(r0, (f4*)(o + c0));
    __builtin_nontemporal_store(r1, (f4*)(o + c1));
  }
}

extern "C" void kernel_launch(void* const* d_in, const int* in_sizes, int n_in,
                              void* d_out, int out_size, void* d_ws, size_t ws_size,
                              hipStream_t stream) {
  const float* rois  = (const float*)d_in[0];
  const float* metas = (const float*)d_in[1];
  const float* fm2   = (const float*)d_in[2];
  const float* fm3   = (const float*)d_in[3];
  const float* fm4   = (const float*)d_in[4];
  const float* fm5   = (const float*)d_in[5];

  const int B = in_sizes[1] / 11;           // img_metas: [B,11]
  const int N = in_sizes[0] / (4 * B);      // rois: [B,N,4]

  dim3 grid(B * N);
  dim3 block(256);
  hipLaunchKernelGGL(roialign_kernel, grid, block, 0, stream,
                     rois, metas, fm2, fm3, fm4, fm5, (float*)d_out, N);
}